// SAS_38809324487317
// MI455X (gfx1250) — compile-verified
//
#include <hip/hip_runtime.h>
#include <hip/hip_bf16.h>

typedef float v2f __attribute__((ext_vector_type(2)));
typedef float v8f __attribute__((ext_vector_type(8)));

#define BSTRIDE 132  // B-tile K-stride (words): 132 % 64 == 4 -> conflict-free b64 frag loads

__device__ __forceinline__ v8f wmma_f32(v2f a, v2f b, v8f c) {
  // D = A(16x4) * B(4x16) + C, full fp32
  return __builtin_amdgcn_wmma_f32_16x16x4_f32(false, a, false, b, (short)0, c, false, false);
}
__device__ __forceinline__ float lrelu(float v) { return v > 0.f ? v : 0.1f * v; }
__device__ __forceinline__ v2f ldsB(const float* Bm, int ncol, int ka) {
  return *(const v2f*)(Bm + ncol * BSTRIDE + ka);  // {B[ka][ncol], B[ka+1][ncol]}
}

// ---------------- prep: fold BN into weights, transpose to [k][co] ----------
__global__ void k_prep(const float* c2w, const float* c2bn,
                       const float* e1w, const float* e1bn,
                       const float* e2w, const float* e2bn,
                       const float* endw, const float* endbn,
                       const float* attw, const float* attbn,
                       float* wT2, float* wTe1, float* wTe2, float* wTend,
                       float* bias2, float* biasE1, float* biasE2, float* biasEnd,
                       float* awS, float* abias) {
  const float EPS = 1e-5f;
  int i0 = blockIdx.x * blockDim.x + threadIdx.x;
  int stride = gridDim.x * blockDim.x;
  // conv2 1x1: (128,128) -> [ci][co]
  for (int i = i0; i < 16384; i += stride) {
    int ci = i >> 7, co = i & 127;
    float inv = c2bn[co] * rsqrtf(c2bn[384 + co] + EPS);
    wT2[i] = c2w[co * 128 + ci] * inv;
  }
  // enc1 3x3: (128,128,3,3) -> [tap][ci][co]
  for (int i = i0; i < 147456; i += stride) {
    int t = i / 16384, r = i - t * 16384, ci = r >> 7, co = r & 127;
    float inv = e1bn[co] * rsqrtf(e1bn[384 + co] + EPS);
    wTe1[i] = e1w[co * 1152 + ci * 9 + t] * inv;
  }
  // enc2 3x3
  for (int i = i0; i < 147456; i += stride) {
    int t = i / 16384, r = i - t * 16384, ci = r >> 7, co = r & 127;
    float inv = e2bn[co] * rsqrtf(e2bn[384 + co] + EPS);
    wTe2[i] = e2w[co * 1152 + ci * 9 + t] * inv;
  }
  // end 1x1: (256,385) -> [ci][co]
  for (int i = i0; i < 98560; i += stride) {
    int ci = i / 256, co = i - ci * 256;
    float inv = endbn[co] * rsqrtf(endbn[768 + co] + EPS);
    wTend[i] = endw[co * 385 + ci] * inv;
  }
  // biases (b - m*inv)
  for (int i = i0; i < 128; i += stride) {
    float iv2 = c2bn[i] * rsqrtf(c2bn[384 + i] + EPS);
    bias2[i] = c2bn[128 + i] - c2bn[256 + i] * iv2;
    float iv1 = e1bn[i] * rsqrtf(e1bn[384 + i] + EPS);
    biasE1[i] = e1bn[128 + i] - e1bn[256 + i] * iv1;
    float ive = e2bn[i] * rsqrtf(e2bn[384 + i] + EPS);
    biasE2[i] = e2bn[128 + i] - e2bn[256 + i] * ive;
  }
  for (int i = i0; i < 256; i += stride) {
    float inv = endbn[i] * rsqrtf(endbn[768 + i] + EPS);
    biasEnd[i] = endbn[256 + i] - endbn[512 + i] * inv;
  }
  // attention conv weights (1,2,7,7) with its 1-channel BN folded
  for (int i = i0; i < 98; i += stride) {
    float invA = attbn[0] * rsqrtf(attbn[3] + EPS);
    awS[i] = attw[i] * invA;
  }
  if (i0 == 0) {
    float invA = attbn[0] * rsqrtf(attbn[3] + EPS);
    abias[0] = attbn[1] - attbn[2] * invA;
  }
}

// ---------------- per-(b,c) spatial mean ------------------------------------
__global__ void k_spatmean(const float* x, float* xmean) {
  __shared__ float red[256];
  int bc = blockIdx.x;
  const float* p = x + (size_t)bc * 6400;
  float s = 0.f;
  for (int i = threadIdx.x; i < 6400; i += 256) s += p[i];
  red[threadIdx.x] = s;
  __syncthreads();
  for (int o = 128; o > 0; o >>= 1) {
    if ((int)threadIdx.x < o) red[threadIdx.x] += red[threadIdx.x + o];
    __syncthreads();
  }
  if (threadIdx.x == 0) xmean[bc] = red[0] * (1.f / 6400.f);
}

// ---------------- ECA 5-tap conv + sigmoid + exact top-128 ------------------
__global__ void k_eca_topk(const float* xmean, const float* ecaw, int* posidx) {
  __shared__ float ym[256], ss[256];
  __shared__ int sel[256];
  int b = blockIdx.x, c = threadIdx.x;
  ym[c] = xmean[b * 256 + c];
  __syncthreads();
  float z = 0.f;
  for (int i = 0; i < 5; i++) {
    int j = c + i - 2;
    if (j >= 0 && j < 256) z += ecaw[i] * ym[j];
  }
  float s = 1.f / (1.f + expf(-z));
  ss[c] = s;
  __syncthreads();
  int rank = 0;  // stable descending rank (matches argsort(-scales))
  for (int j = 0; j < 256; j++) {
    float sj = ss[j];
    rank += (sj > s) || (sj == s && j < c);
  }
  sel[c] = (rank < 128) ? 1 : 0;
  __syncthreads();
  if (sel[c]) {
    int pos = 0;
    for (int j = 0; j < c; j++) pos += sel[j];
    posidx[b * 128 + pos] = c;  // ascending channel order == jnp.sort(top_idx)
  }
}

// ---------------- per-pixel channel sum/mean/max ----------------------------
__global__ void k_chanstats(const float* in, int C, int HW, int n,
                            float* osum, float* omean, float* omax) {
  for (int i = blockIdx.x * blockDim.x + threadIdx.x; i < n;
       i += gridDim.x * blockDim.x) {
    int b = i / HW, p = i - b * HW;
    const float* base = in + (size_t)b * C * HW + p;
    float s = 0.f, mx = -3.4e38f;
    for (int c = 0; c < C; c++) {
      float v = base[(size_t)c * HW];
      s += v;
      mx = fmaxf(mx, v);
    }
    if (osum) osum[i] = s;
    omean[i] = s / (float)C;
    omax[i] = mx;
  }
}

// ---------------- 7x7 attention map: sigmoid(lrelu(conv+bias))+1 ------------
__global__ void k_att(const float* cm, const float* cx, int shiftSrc, int srcW,
                      int srcHW, const float* a1, const float* awS,
                      const float* abias, float* out, int n) {
  for (int i = blockIdx.x * blockDim.x + threadIdx.x; i < n;
       i += gridDim.x * blockDim.x) {
    int b = i / 6400, p = i - b * 6400, y = p / 80, x = p - y * 80;
    float acc = abias[0];
    for (int ch = 0; ch < 2; ch++) {
      const float* src = ch ? cx : cm;
      const float* wb = awS + ch * 49;
      for (int kh = 0; kh < 7; kh++) {
        int yy = y + kh - 3;
        if ((unsigned)yy >= 80u) continue;
        for (int kw = 0; kw < 7; kw++) {
          int xx = x + kw - 3;
          if ((unsigned)xx >= 80u) continue;
          int si = shiftSrc ? ((yy >> 1) * srcW + (xx >> 1)) : (yy * srcW + xx);
          float v = src[(size_t)b * srcHW + si];
          if (a1) v *= a1[(size_t)b * 6400 + yy * 80 + xx];
          acc += wb[kh * 7 + kw] * v;
        }
      }
    }
    float zz = lrelu(acc);
    out[i] = 1.f + 1.f / (1.f + expf(-zz));
  }
}

// ---------------- conv2 1x1 (128->128) via WMMA, 64-pixel tiles -------------
// B tile staged column-major: Bm[pixel][k] -> one ds_load_b64 per fragment
__global__ void __launch_bounds__(256) k_conv2(const float* x, const float* wT2,
                                               const float* bias2, float* tmp1) {
  __shared__ __align__(16) float Bm[64 * BSTRIDE];
  int b = blockIdx.y, pbase = blockIdx.x * 64;
  const float* xin = x + (size_t)b * 256 * 6400;  // split1 = channels 0..127
  for (int idx = threadIdx.x; idx < 128 * 64; idx += 256) {
    int ci = idx >> 6, j = idx & 63;  // j fastest -> coalesced global reads
    Bm[j * BSTRIDE + ci] = xin[(size_t)ci * 6400 + pbase + j];
  }
  __syncthreads();
  int lane = threadIdx.x & 31, wave = threadIdx.x >> 5;
  int half = lane >> 4, l16 = lane & 15, cobase = wave * 16;
  v8f acc[4] = {};
  for (int k = 0; k < 128; k += 4) {
    int ka = k + 2 * half;
    v2f a;
    a.x = wT2[ka * 128 + cobase + l16];
    a.y = wT2[(ka + 1) * 128 + cobase + l16];
#pragma unroll
    for (int nt = 0; nt < 4; nt++) {
      v2f bb = ldsB(Bm, nt * 16 + l16, ka);
      acc[nt] = wmma_f32(a, bb, acc[nt]);
    }
  }
#pragma unroll
  for (int nt = 0; nt < 4; nt++)
#pragma unroll
    for (int r = 0; r < 8; r++) {
      int co = cobase + r + 8 * half;
      float v = acc[nt][r] + bias2[co];
      tmp1[((size_t)b * 128 + co) * 6400 + pbase + nt * 16 + l16] = lrelu(v);
    }
}

// ---------------- 3x3 s2 p1 conv (128->128) via per-tap im2col + WMMA -------
// srcHalf=0: input = x channels 128..255 on 80x80 (enc1)
// srcHalf=1: input = e1 on 40x40, read through nearest up2 (enc2)
// 32 output pixels per block (2 N-tiles per wave)
__global__ void __launch_bounds__(256) k_enc(const float* in, int srcHalf,
                                             const float* wT, const float* bias,
                                             float* out) {
  __shared__ __align__(16) float Bm[32 * BSTRIDE];
  int b = blockIdx.y, pbase = blockIdx.x * 32;
  int lane = threadIdx.x & 31, wave = threadIdx.x >> 5;
  int half = lane >> 4, l16 = lane & 15, cobase = wave * 16;
  v8f acc[2] = {};
  for (int t = 0; t < 9; t++) {
    int kh = t / 3, kw = t - kh * 3;
    __syncthreads();
    for (int idx = threadIdx.x; idx < 128 * 32; idx += 256) {
      int ci = idx >> 5, j = idx & 31;
      int p = pbase + j, oy = p / 40, ox = p - oy * 40;
      int iy = 2 * oy - 1 + kh, ix = 2 * ox - 1 + kw;
      float v = 0.f;
      if ((unsigned)iy < 80u && (unsigned)ix < 80u) {
        v = srcHalf ? in[((size_t)b * 128 + ci) * 1600 + (iy >> 1) * 40 + (ix >> 1)]
                    : in[((size_t)b * 256 + 128 + ci) * 6400 + iy * 80 + ix];
      }
      Bm[j * BSTRIDE + ci] = v;
    }
    __syncthreads();
    const float* wA = wT + t * 16384;
    for (int k = 0; k < 128; k += 4) {
      int ka = k + 2 * half;
      v2f a;
      a.x = wA[ka * 128 + cobase + l16];
      a.y = wA[(ka + 1) * 128 + cobase + l16];
#pragma unroll
      for (int nt = 0; nt < 2; nt++) {
        v2f bb = ldsB(Bm, nt * 16 + l16, ka);
        acc[nt] = wmma_f32(a, bb, acc[nt]);
      }
    }
  }
#pragma unroll
  for (int nt = 0; nt < 2; nt++)
#pragma unroll
    for (int r = 0; r < 8; r++) {
      int co = cobase + r + 8 * half;
      out[((size_t)b * 128 + co) * 1600 + pbase + nt * 16 + l16] =
          lrelu(acc[nt][r] + bias[co]);
    }
}

// ---------------- final 1x1 conv (385->256) via WMMA, K in 3 LDS chunks -----
__global__ void __launch_bounds__(512) k_end(const float* x, const float* tmp1,
                                             const float* e2, const float* a2,
                                             const float* a3, const int* posidx,
                                             const float* chsum,
                                             const float* wTend,
                                             const float* biasEnd, float* out) {
  __shared__ __align__(16) float Bm[64 * BSTRIDE];
  __shared__ float mn_s[64];
  __shared__ int pos_s[128];
  int b = blockIdx.y, pbase = blockIdx.x * 64;
  int tid = threadIdx.x;
  if (tid < 128) pos_s[tid] = posidx[b * 128 + tid];
  int lane = tid & 31, wave = tid >> 5;
  int half = lane >> 4, l16 = lane & 15, cobase = wave * 16;  // 16 waves -> co 0..255
  v8f acc[4] = {};
  for (int chunk = 0; chunk < 3; chunk++) {
    __syncthreads();
    for (int idx = tid; idx < 128 * 64; idx += 512) {
      int ci = idx >> 6, j = idx & 63, p = pbase + j;
      float v;
      if (chunk == 0) {
        v = tmp1[((size_t)b * 128 + ci) * 6400 + p] * a2[(size_t)b * 6400 + p];
      } else if (chunk == 1) {
        int y = p / 80, xx = p - y * 80;
        v = e2[((size_t)b * 128 + ci) * 1600 + (y >> 1) * 40 + (xx >> 1)] *
            a3[(size_t)b * 6400 + p];
      } else {
        v = x[((size_t)b * 256 + pos_s[ci]) * 6400 + p];
      }
      Bm[j * BSTRIDE + ci] = v;
    }
    __syncthreads();
    if (chunk == 2) {
      if (tid < 64) {  // mean of the 128 non-selected channels, per pixel
        float s = 0.f;
        for (int ci = 0; ci < 128; ci++) s += Bm[tid * BSTRIDE + ci];
        mn_s[tid] = (chsum[(size_t)b * 6400 + pbase + tid] - s) * (1.f / 128.f);
      }
      __syncthreads();
    }
    const float* wA = wTend + chunk * 128 * 256;
    for (int k = 0; k < 128; k += 4) {
      int ka = k + 2 * half;
      v2f a;
      a.x = wA[ka * 256 + cobase + l16];
      a.y = wA[(ka + 1) * 256 + cobase + l16];
#pragma unroll
      for (int nt = 0; nt < 4; nt++) {
        v2f bb = ldsB(Bm, nt * 16 + l16, ka);
        acc[nt] = wmma_f32(a, bb, acc[nt]);
      }
    }
  }
  const float* w384 = wTend + 384 * 256;  // mean_nega row as rank-1 update
#pragma unroll
  for (int nt = 0; nt < 4; nt++)
#pragma unroll
    for (int r = 0; r < 8; r++) {
      int co = cobase + r + 8 * half;
      float v = acc[nt][r] + w384[co] * mn_s[nt * 16 + l16] + biasEnd[co];
      out[((size_t)b * 256 + co) * 6400 + pbase + nt * 16 + l16] = lrelu(v);
    }
}

extern "C" void kernel_launch(void* const* d_in, const int* in_sizes, int n_in,
                              void* d_out, int out_size, void* d_ws,
                              size_t ws_size, hipStream_t stream) {
  (void)in_sizes; (void)n_in; (void)out_size; (void)ws_size;
  const float* x = (const float*)d_in[0];
  const float* ecaw = (const float*)d_in[1];
  const float* attw = (const float*)d_in[2];
  const float* attbn = (const float*)d_in[3];
  const float* c2w = (const float*)d_in[4];
  const float* c2bn = (const float*)d_in[5];
  const float* e1w = (const float*)d_in[6];
  const float* e1bn = (const float*)d_in[7];
  const float* e2w = (const float*)d_in[8];
  const float* e2bn = (const float*)d_in[9];
  const float* endw = (const float*)d_in[10];
  const float* endbn = (const float*)d_in[11];

  float* ws = (float*)d_ws;
  size_t off = 0;
  auto alloc = [&](size_t n) {
    float* p = ws + off;
    off += (n + 63) & ~(size_t)63;
    return p;
  };
  float* WT2 = alloc(16384);
  float* WTE1 = alloc(147456);
  float* WTE2 = alloc(147456);
  float* WTEND = alloc(98560);
  float* B2 = alloc(128);
  float* BE1 = alloc(128);
  float* BE2 = alloc(128);
  float* BEND = alloc(256);
  float* AWS = alloc(128);
  float* ABIAS = alloc(64);
  float* XMEAN = alloc(4096);
  int* POS = (int*)alloc(2048);
  float* CHSUM = alloc(102400);
  float* CHMEAN = alloc(102400);
  float* CHMAX = alloc(102400);
  float* TMP1 = alloc(13107200);
  float* E1 = alloc(3276800);
  float* E2 = alloc(3276800);
  float* M1 = alloc(102400);
  float* MX1 = alloc(102400);
  float* M2 = alloc(25600);
  float* MX2 = alloc(25600);
  float* A1 = alloc(102400);
  float* A2 = alloc(102400);
  float* A3 = alloc(102400);

  k_prep<<<1024, 256, 0, stream>>>(c2w, c2bn, e1w, e1bn, e2w, e2bn, endw, endbn,
                                   attw, attbn, WT2, WTE1, WTE2, WTEND, B2, BE1,
                                   BE2, BEND, AWS, ABIAS);
  k_spatmean<<<4096, 256, 0, stream>>>(x, XMEAN);
  k_eca_topk<<<16, 256, 0, stream>>>(XMEAN, ecaw, POS);
  k_chanstats<<<400, 256, 0, stream>>>(x, 256, 6400, 102400, CHSUM, CHMEAN, CHMAX);
  k_conv2<<<dim3(100, 16), 256, 0, stream>>>(x, WT2, B2, TMP1);
  k_enc<<<dim3(50, 16), 256, 0, stream>>>(x, 0, WTE1, BE1, E1);
  k_enc<<<dim3(50, 16), 256, 0, stream>>>(E1, 1, WTE2, BE2, E2);
  k_chanstats<<<400, 256, 0, stream>>>(TMP1, 128, 6400, 102400, nullptr, M1, MX1);
  k_chanstats<<<100, 256, 0, stream>>>(E2, 128, 1600, 25600, nullptr, M2, MX2);
  k_att<<<400, 256, 0, stream>>>(CHMEAN, CHMAX, 0, 80, 6400, nullptr, AWS, ABIAS,
                                 A1, 102400);
  k_att<<<400, 256, 0, stream>>>(M1, MX1, 0, 80, 6400, A1, AWS, ABIAS, A2, 102400);
  k_att<<<400, 256, 0, stream>>>(M2, MX2, 1, 40, 1600, A1, AWS, ABIAS, A3, 102400);
  k_end<<<dim3(100, 16), 512, 0, stream>>>(x, TMP1, E2, A2, A3, POS, CHSUM,
                                           WTEND, BEND, (float*)d_out);
}